// WordKVMN_58695023067808
// MI455X (gfx1250) — compile-verified
//
#include <hip/hip_runtime.h>
#include <hip/hip_bf16.h>

// Problem constants (from reference)
#define BB 4
#define LL 256
#define WW 128
#define HH 768
#define NLAB 10
#define INV_TEMPER 0.03608439182435161f   // 1/sqrt(768)

typedef __attribute__((ext_vector_type(2))) float v2f;
typedef __attribute__((ext_vector_type(4))) float v4f;
typedef __attribute__((ext_vector_type(8))) float v8f;

// -----------------------------------------------------------------------------
// Kernel 1: u = hidden . emb_a[word_seq]^T  (WMMA f32 16x16x4),
//           masked-exp normalize over W, bin p by label -> s[B,L,10] (workspace)
// grid: (L/16, B), block: 256 (8 waves; wave i owns W-tile i of 16 columns)
// -----------------------------------------------------------------------------
__global__ __launch_bounds__(256) void wordkvmn_attn_bin(
    const int* __restrict__ word_seq,      // [B,W]
    const float* __restrict__ hidden,      // [B,L,H]
    const int* __restrict__ labels,        // [B,L,W]
    const float* __restrict__ wmask,       // [B,L,W]
    const float* __restrict__ emb_a,       // [WORD_SIZE,H]
    float* __restrict__ s_ws)              // [B,L,10]
{
    const int ltile = blockIdx.x;          // 0..15
    const int b     = blockIdx.y;          // 0..3
    const int l0    = ltile * 16;
    const int tid   = threadIdx.x;
    const int wave  = tid >> 5;            // 0..7  -> w-tile
    const int lane  = tid & 31;

    __shared__ float lds_A[16 * HH];       // 48 KB: hidden tile, row-major [m][k]
    __shared__ float lds_delta[16 * WW];   // 8 KB : exp(u)*mask
    __shared__ float lds_rowsum[16];
    __shared__ float lds_s[16 * NLAB];
    __shared__ int   lds_widx[WW];

    // ---- stage hidden tile (16 x 768 f32) into LDS, coalesced b128 ----
    const float* Ag = hidden + ((size_t)(b * LL + l0)) * HH;
    for (int i = tid; i < 16 * HH / 4; i += 256)
        ((v4f*)lds_A)[i] = ((const v4f*)Ag)[i];
    if (tid < WW)        lds_widx[tid]   = word_seq[b * WW + tid];
    if (tid < 16)        lds_rowsum[tid] = 0.0f;
    if (tid < 16 * NLAB) lds_s[tid]      = 0.0f;
    __syncthreads();

    // ---- WMMA accumulation: C[16l x 16w] over K = H in steps of 4 ----
    const int w0    = wave * 16;
    const int n     = lane & 15;                  // A: row m ; B: col n (w)
    const int khalf = (lane < 16) ? 0 : 2;        // f32 16x16x4 operand layout
    const float* apnt = lds_A + n * HH;           // A row m = lane&15
    const size_t arow = (size_t)lds_widx[w0 + n]; // B column n -> emb_a row
    const float* bpnt = emb_a + arow * HH;

    v8f acc = (v8f)0.0f;
    for (int k0 = 0; k0 < HH; k0 += 4) {
        v2f av = *(const v2f*)(apnt + k0 + khalf);    // ds_load_b64
        v2f bv = *(const v2f*)(bpnt + k0 + khalf);    // global_load_b64 (L2 hit)
        acc = __builtin_amdgcn_wmma_f32_16x16x4_f32(
                  false, av, false, bv, (short)0, acc, false, false);
    }

    // ---- delta = exp(u/temper) * clip(mask,0,1), partial row sums ----
    float rsum[8];
    #pragma unroll
    for (int i = 0; i < 8; i++) {
        const int m = i + ((lane < 16) ? 0 : 8);
        const int w = w0 + (lane & 15);
        float u  = acc[i] * INV_TEMPER;
        float mk = wmask[((size_t)(b * LL + l0 + m)) * WW + w];
        mk = fminf(fmaxf(mk, 0.0f), 1.0f);
        float dl = __expf(u) * mk;
        lds_delta[m * WW + w] = dl;
        rsum[i] = dl;
    }
    // butterfly over the 16 lanes of each half (rows stay within half-wave)
    #pragma unroll
    for (int i = 0; i < 8; i++) {
        #pragma unroll
        for (int off = 1; off < 16; off <<= 1)
            rsum[i] += __shfl_xor(rsum[i], off, 32);
    }
    if ((lane & 15) == 0) {
        #pragma unroll
        for (int i = 0; i < 8; i++) {
            const int m = i + ((lane < 16) ? 0 : 8);
            atomicAdd(&lds_rowsum[m], rsum[i]);       // ds_add_f32
        }
    }
    __syncthreads();

    // ---- normalize and bin by label: s[m][lbl] += p ----
    {
        const int r    = tid >> 4;          // 0..15
        const int wseg = (tid & 15) * 8;    // 8 columns per thread
        const float inv = 1.0f / (lds_rowsum[r] + 1e-10f);
        const size_t lrow = ((size_t)(b * LL + l0 + r)) * WW;
        #pragma unroll
        for (int j = 0; j < 8; j++) {
            const int w = wseg + j;
            float p = lds_delta[r * WW + w] * inv;
            int lbl = labels[lrow + w];
            atomicAdd(&lds_s[r * NLAB + lbl], p);
        }
    }
    __syncthreads();

    if (tid < 16 * NLAB)
        s_ws[(size_t)(b * LL + l0) * NLAB + tid] = lds_s[tid];
}

// -----------------------------------------------------------------------------
// Kernel 2 (memory-bound, ~805 MB of row gathers):
//   out[b,l,:] = hidden[b,l,:] + sum_j s[b,l,j]*emb_c[j,:]
//              + sum_w emb_d[freq[b,l,w],:] + sum_w emb_e[avfreq[b,l,w],:]
// grid: (L, B), block: 192 (each thread owns 4 H-columns; 192*4 = 768)
// -----------------------------------------------------------------------------
__global__ __launch_bounds__(192) void wordkvmn_gather_sum(
    const float* __restrict__ hidden,      // [B,L,H]
    const int* __restrict__ freq,          // [B,L,W]
    const int* __restrict__ avfreq,        // [B,L,W]
    const float* __restrict__ emb_c,       // [10,H]
    const float* __restrict__ emb_d,       // [WORD_SIZE,H]
    const float* __restrict__ emb_e,       // [WORD_SIZE,H]
    const float* __restrict__ s_ws,        // [B,L,10]
    float* __restrict__ out)               // [B,L,H]
{
    const int l   = blockIdx.x;
    const int b   = blockIdx.y;
    const int tid = threadIdx.x;
    const size_t row = (size_t)b * LL + l;

    __shared__ int   fidx[WW];
    __shared__ int   aidx[WW];
    __shared__ float sv[NLAB];

    for (int i = tid; i < WW; i += 192) {
        fidx[i] = freq[row * WW + i];
        aidx[i] = avfreq[row * WW + i];
    }
    if (tid < NLAB) sv[tid] = s_ws[row * NLAB + tid];
    __syncthreads();

    const int h = tid * 4;
    v4f acc = *(const v4f*)(hidden + row * HH + h);   // residual

    // tiny [1x10] x [10xH] contraction, emb_c is 30 KB -> L2/L0 resident
    #pragma unroll
    for (int j = 0; j < NLAB; j++) {
        v4f c4 = *(const v4f*)(emb_c + (size_t)j * HH + h);
        acc += sv[j] * c4;
    }

    // prime software prefetch for the first two random rows
    __builtin_prefetch(emb_d + (size_t)fidx[0] * HH + h, 0, 1);
    __builtin_prefetch(emb_e + (size_t)aidx[0] * HH + h, 0, 1);
    __builtin_prefetch(emb_d + (size_t)fidx[1] * HH + h, 0, 1);
    __builtin_prefetch(emb_e + (size_t)aidx[1] * HH + h, 0, 1);

    for (int w = 0; w < WW; w++) {
        if (w + 2 < WW) {   // global_prefetch_b8: hide latency of random rows
            __builtin_prefetch(emb_d + (size_t)fidx[w + 2] * HH + h, 0, 1);
            __builtin_prefetch(emb_e + (size_t)aidx[w + 2] * HH + h, 0, 1);
        }
        v4f d4 = *(const v4f*)(emb_d + (size_t)fidx[w] * HH + h);  // coalesced b128
        v4f e4 = *(const v4f*)(emb_e + (size_t)aidx[w] * HH + h);
        acc += d4 + e4;
    }

    *(v4f*)(out + row * HH + h) = acc;
}

// -----------------------------------------------------------------------------
extern "C" void kernel_launch(void* const* d_in, const int* in_sizes, int n_in,
                              void* d_out, int out_size, void* d_ws, size_t ws_size,
                              hipStream_t stream) {
    const int*   word_seq = (const int*)  d_in[0];
    const float* hidden   = (const float*)d_in[1];
    const int*   labels   = (const int*)  d_in[2];
    const float* wmask    = (const float*)d_in[3];
    const int*   freq     = (const int*)  d_in[4];
    const int*   avfreq   = (const int*)  d_in[5];
    const float* emb_a    = (const float*)d_in[6];
    const float* emb_c    = (const float*)d_in[7];
    const float* emb_d    = (const float*)d_in[8];
    const float* emb_e    = (const float*)d_in[9];
    float*       out      = (float*)d_out;
    float*       s_ws     = (float*)d_ws;           // needs B*L*10*4 = 40 KB

    dim3 g1(LL / 16, BB);
    wordkvmn_attn_bin<<<g1, dim3(256), 0, stream>>>(
        word_seq, hidden, labels, wmask, emb_a, s_ws);

    dim3 g2(LL, BB);
    wordkvmn_gather_sum<<<g2, dim3(192), 0, stream>>>(
        hidden, freq, avfreq, emb_c, emb_d, emb_e, s_ws, out);
}